// CommunityPassing_90108413870708
// MI455X (gfx1250) — compile-verified
//
#include <hip/hip_runtime.h>
#include <hip/hip_bf16.h>

#define NUM_COMMS 1000
#define THREADS   256
#define FCHUNK    32            // floats of F handled per block (grid.y)
#define LDS_STRIDE 33           // pad so bank = (33*c + f) % 64 spreads communities
#define NCHUNKS   64            // node chunks (grid.x)

// Async global->LDS support (device pass only; host pass sees fallback).
#if defined(__has_builtin)
#if __has_builtin(__builtin_amdgcn_global_load_async_to_lds_b128)
#define HAVE_ASYNC_LDS 1
typedef int v4i __attribute__((__vector_size__(16)));
typedef __attribute__((address_space(1))) v4i gv4i;
typedef __attribute__((address_space(3))) v4i lv4i;
#if __has_builtin(__builtin_amdgcn_s_wait_asynccnt)
#define ASYNC_WAIT(n) __builtin_amdgcn_s_wait_asynccnt(n)
#else
#define ASYNC_WAIT(n) asm volatile("s_wait_asynccnt " #n ::: "memory")
#endif
#endif
#endif

// Stage: 2 buffers x (64 nodes x 8 float4) = 1024 float4 = 16 KB
#define STAGE_F4   512
#define STAGE_BYTES (2 * STAGE_F4 * 16)
#define ACC_FLOATS (NUM_COMMS * LDS_STRIDE)

// ---------------------------------------------------------------------------
// Zero the workspace: comm_sum [NUM_COMMS*F] + cnt [NUM_COMMS]
// ---------------------------------------------------------------------------
__global__ void k_zero(float* __restrict__ ws, int total) {
    int i = blockIdx.x * blockDim.x + threadIdx.x;
    if (i < total) ws[i] = 0.0f;
}

// ---------------------------------------------------------------------------
// Per-community counts: LDS histogram per block, then global atomic flush
// ---------------------------------------------------------------------------
__global__ void k_count(const int* __restrict__ comm, float* __restrict__ gcnt,
                        int N, int chunk) {
    __shared__ float h[NUM_COMMS];
    for (int i = threadIdx.x; i < NUM_COMMS; i += THREADS) h[i] = 0.0f;
    __syncthreads();
    int start = blockIdx.x * chunk;
    int end   = start + chunk; if (end > N) end = N;
    for (int n = start + threadIdx.x; n < end; n += THREADS)
        atomicAdd(&h[comm[n]], 1.0f);            // ds_add_f32 (no-rtn)
    __syncthreads();
    for (int i = threadIdx.x; i < NUM_COMMS; i += THREADS) {
        float v = h[i];
        if (v != 0.0f) atomicAdd(&gcnt[i], v);   // global_atomic_add_f32
    }
}

// ---------------------------------------------------------------------------
// Scatter-add x into per-community sums.
// grid = (NCHUNKS, F/FCHUNK). Each block owns a 1000 x 32 fp32 LDS accumulator
// (stride 33: bank = (33c+f)%64 spreads communities) plus a 16 KB stage.
// Per-wave async double-buffer: each thread issues async b128 loads for
// elements e=tid and e=tid+256 of the next tile, then consumes exactly those
// elements of the current tile -> no cross-wave sync, only s_wait_asynccnt.
// ---------------------------------------------------------------------------
__global__ void k_scatter(const float4* __restrict__ x4,
                          const int* __restrict__ comm,
                          float* __restrict__ gsum,
                          int N, int chunk, int F) {
    extern __shared__ float dyn[];
    float* acc = dyn;                            // ACC_FLOATS
    const int fbase  = blockIdx.y * FCHUNK;
    const int f4row  = F >> 2;                   // float4 per row (=64)
    const int f4base = fbase >> 2;
    const int q  = threadIdx.x & 7;              // float4 slot within F-chunk
    const int ni = threadIdx.x >> 3;             // node sub-index 0..31

    for (int i = threadIdx.x; i < ACC_FLOATS; i += THREADS) acc[i] = 0.0f;
    __syncthreads();

    int start = blockIdx.x * chunk;
    int end   = start + chunk; if (end > N) end = N;

#if defined(HAVE_ASYNC_LDS)
    float4* stage4 = (float4*)(dyn + ACC_FLOATS);    // 2 * STAGE_F4
    const int e0     = threadIdx.x;
    const int ntiles = (end - start) >> 6;           // full 64-node tiles

    auto issue_tile = [&](int t, int nb) {
        const int n0 = start + (t << 6);
        const float4* g0 = &x4[(size_t)(n0 + ni) * f4row + f4base + q];
        const float4* g1 = &x4[(size_t)(n0 + ni + 32) * f4row + f4base + q];
        __builtin_amdgcn_global_load_async_to_lds_b128(
            (gv4i*)g0, (lv4i*)&stage4[(nb << 9) + e0], 0, 0);
        __builtin_amdgcn_global_load_async_to_lds_b128(
            (gv4i*)g1, (lv4i*)&stage4[(nb << 9) + e0 + 256], 0, 0);
    };

    if (ntiles > 0) issue_tile(0, 0);
    for (int t = 0; t < ntiles; ++t) {
        const int nb = t & 1;
        if (t + 1 < ntiles) { issue_tile(t + 1, nb ^ 1); ASYNC_WAIT(2); }
        else                { ASYNC_WAIT(0); }
        const int n0 = start + (t << 6);
        int    c0 = comm[n0 + ni];
        int    c1 = comm[n0 + ni + 32];
        float4 v0 = stage4[(nb << 9) + e0];          // ds_load_b128
        float4 v1 = stage4[(nb << 9) + e0 + 256];
        float* a0 = &acc[c0 * LDS_STRIDE + (q << 2)];
        atomicAdd(&a0[0], v0.x); atomicAdd(&a0[1], v0.y);
        atomicAdd(&a0[2], v0.z); atomicAdd(&a0[3], v0.w);
        float* a1 = &acc[c1 * LDS_STRIDE + (q << 2)];
        atomicAdd(&a1[0], v1.x); atomicAdd(&a1[1], v1.y);
        atomicAdd(&a1[2], v1.z); atomicAdd(&a1[3], v1.w);
    }
    const int tail_start = start + (ntiles << 6);
#else
    const int tail_start = start;
#endif

    // tail (and full fallback if async builtins are unavailable)
    for (int n = tail_start + ni; n < end; n += 32) {
        int    c = comm[n];
        float4 v = x4[(size_t)n * f4row + f4base + q];
        float* a = &acc[c * LDS_STRIDE + (q << 2)];
        atomicAdd(&a[0], v.x); atomicAdd(&a[1], v.y);
        atomicAdd(&a[2], v.z); atomicAdd(&a[3], v.w);
    }
    __syncthreads();

    for (int i = threadIdx.x; i < NUM_COMMS * FCHUNK; i += THREADS) {
        int   c = i >> 5;                        // FCHUNK == 32
        int   f = i & 31;
        float v = acc[c * LDS_STRIDE + f];
        if (v != 0.0f)
            atomicAdd(&gsum[c * F + fbase + f], v);
    }
}

// ---------------------------------------------------------------------------
// comm_mean = comm_sum / max(cnt, eps), in place (1000*256 elements)
// ---------------------------------------------------------------------------
__global__ void k_norm(float* __restrict__ gsum, const float* __restrict__ gcnt,
                       int total, int F) {
    int i = blockIdx.x * blockDim.x + threadIdx.x;
    if (i < total) {
        int c = i / F;
        gsum[i] = gsum[i] / fmaxf(gcnt[c], 1e-12f);
    }
}

// ---------------------------------------------------------------------------
// Gather: out[n, :] = comm_mean[community[n], :], float4 vectorized.
// Reads hit the L2-resident 1 MB table; writes are streaming b128.
// ---------------------------------------------------------------------------
__global__ void k_gather(const float4* __restrict__ mean4,
                         const int* __restrict__ comm,
                         float4* __restrict__ out4,
                         long long total4) {
    long long i = (long long)blockIdx.x * THREADS + threadIdx.x;
    if (i < total4) {
        int n = (int)(i >> 6);                   // F/4 == 64
        int p = (int)(i & 63);
        int c = comm[n];
        out4[i] = mean4[(size_t)c * 64 + p];
    }
}

// ---------------------------------------------------------------------------
extern "C" void kernel_launch(void* const* d_in, const int* in_sizes, int n_in,
                              void* d_out, int out_size, void* d_ws, size_t ws_size,
                              hipStream_t stream) {
    const float* x    = (const float*)d_in[0];
    const int*   comm = (const int*)d_in[1];
    float*       out  = (float*)d_out;

    const int N = in_sizes[1];                 // 500000
    const int F = in_sizes[0] / N;             // 256

    float* gsum = (float*)d_ws;                // NUM_COMMS * F
    float* gcnt = gsum + (size_t)NUM_COMMS * F;

    const int ws_total = NUM_COMMS * F + NUM_COMMS;
    k_zero<<<(ws_total + THREADS - 1) / THREADS, THREADS, 0, stream>>>(gsum, ws_total);

    const int chunk = (N + NCHUNKS - 1) / NCHUNKS;
    k_count<<<NCHUNKS, THREADS, 0, stream>>>(comm, gcnt, N, chunk);

    dim3 sg(NCHUNKS, F / FCHUNK);              // 64 x 8
    // Allocate stage bytes unconditionally: host pass can't see the device
    // __has_builtin result, but the device binary always uses the stage.
    size_t lds_bytes = (size_t)ACC_FLOATS * sizeof(float) + STAGE_BYTES; // ~148 KB
    k_scatter<<<sg, THREADS, lds_bytes, stream>>>((const float4*)x, comm, gsum,
                                                  N, chunk, F);

    const int tot = NUM_COMMS * F;
    k_norm<<<(tot + THREADS - 1) / THREADS, THREADS, 0, stream>>>(gsum, gcnt, tot, F);

    long long total4 = (long long)N * (F / 4);
    long long blocks = (total4 + THREADS - 1) / THREADS;
    k_gather<<<(unsigned)blocks, THREADS, 0, stream>>>((const float4*)gsum, comm,
                                                       (float4*)out, total4);
}